// AMYGNN_67422396613062
// MI455X (gfx1250) — compile-verified
//
#include <hip/hip_runtime.h>
#include <math.h>

typedef __attribute__((ext_vector_type(2))) float v2f;
typedef __attribute__((ext_vector_type(8))) float v8f;

#define N_NODES 100000
#define N_GRAPHS 2000
#define NPG 50
#define THREADS 256

// ---------------------------------------------------------------- utilities
__global__ void zero_kernel(int* counts, int* cursor, float* stats) {
    int i = blockIdx.x * blockDim.x + threadIdx.x;
    if (i < N_GRAPHS) { counts[i] = 0; cursor[i] = 0; }
    if (i < 3 * 128) stats[i] = 0.0f;
}

__global__ void count_kernel(const int* __restrict__ src, int* counts, int E) {
    int e = blockIdx.x * blockDim.x + threadIdx.x;
    if (e < E) atomicAdd(&counts[src[e] / NPG], 1);
}

__global__ __launch_bounds__(1024) void scan_kernel(const int* __restrict__ counts,
                                                    int* __restrict__ offs) {
    __shared__ int cnt[N_GRAPHS];
    __shared__ int pre[N_GRAPHS + 1];
    int t = threadIdx.x;
    for (int i = t; i < N_GRAPHS; i += 1024) cnt[i] = counts[i];
    __syncthreads();
    if (t == 0) {
        int acc = 0;
        for (int i = 0; i < N_GRAPHS; ++i) { pre[i] = acc; acc += cnt[i]; }
        pre[N_GRAPHS] = acc;
    }
    __syncthreads();
    for (int i = t; i <= N_GRAPHS; i += 1024) offs[i] = pre[i];
}

__global__ void scatter_kernel(const int* __restrict__ src, const int* __restrict__ offs,
                               int* cursor, int* __restrict__ esort, int E) {
    int e = blockIdx.x * blockDim.x + threadIdx.x;
    if (e < E) {
        int g = src[e] / NPG;
        int pos = atomicAdd(&cursor[g], 1);
        esort[offs[g] + pos] = e;
    }
}

// ------------------------------------------------- fused GraphConv+ReLU layer
// one block per graph; LDS-resident 50x Fin tile; edge agg via ds_add_f32;
// GEMM (agg@Wrel + x@Wroot + b) via V_WMMA_F32_16X16X4_F32; BN partial sums.
template <int Fin, int Fout, bool BN_IN>
__global__ __launch_bounds__(THREADS) void layer_kernel(
    const float* __restrict__ xin, const float* __restrict__ inscale,
    const float* __restrict__ inshift, const int* __restrict__ src,
    const int* __restrict__ dst, const float* __restrict__ ew,
    const int* __restrict__ offs, const int* __restrict__ esort,
    const float* __restrict__ wrel, const float* __restrict__ brel,
    const float* __restrict__ wroot, float* __restrict__ hout,
    float* __restrict__ stats) {
    constexpr int ROWS = 64;          // 50 rows padded to 4 WMMA tiles
    constexpr int LDP = Fin + 4;      // bank-conflict padding, keeps 16B rows
    __shared__ float xs[ROWS * LDP];
    __shared__ float ag[ROWS * LDP];
    __shared__ float bsum[Fout];
    __shared__ float bsq[Fout];

    const int g = blockIdx.x;
    const int base = g * NPG;
    const int tid = threadIdx.x;
    const int wave = tid >> 5, lane = tid & 31;

    for (int i = tid; i < ROWS * Fin; i += THREADS) {
        int r = i / Fin, c = i - r * Fin;
        float v = 0.0f;
        if (r < NPG) {
            v = xin[(size_t)(base + r) * Fin + c];
            if (BN_IN) v = v * inscale[c] + inshift[c];
        }
        xs[r * LDP + c] = v;
        ag[r * LDP + c] = 0.0f;
    }
    if (tid < Fout) { bsum[tid] = 0.0f; bsq[tid] = 0.0f; }
    __syncthreads();

    // weighted edge aggregation: wave-per-edge, lane-per-feature, LDS atomics
    const int e0 = offs[g], e1 = offs[g + 1];
    constexpr int FPL = Fin / 32;
    for (int e = e0 + wave; e < e1; e += THREADS / 32) {
        int eid = esort[e];
        int s = src[eid] - base;
        int d = dst[eid] - base;
        float w = ew[eid];
#pragma unroll
        for (int j = 0; j < FPL; ++j) {
            int f = lane + 32 * j;
            atomicAdd(&ag[d * LDP + f], xs[s * LDP + f] * w);
        }
    }
    __syncthreads();

    // dual GEMM on matrix pipes: C = b_rel + agg@Wrel + x@Wroot  (fp32 WMMA)
    constexpr int NT = Fout / 16;
    const int half = lane >> 4, lr = lane & 15;
    for (int t = wave; t < 4 * NT; t += THREADS / 32) {  // wave-uniform: EXEC full
        int m0 = (t / NT) * 16;
        int n0 = (t % NT) * 16;
        int n = n0 + lr;
        float bias = brel[n];
        v8f c;
#pragma unroll
        for (int r = 0; r < 8; ++r) c[r] = bias;
        const float* Arow = &ag[(m0 + lr) * LDP];
        const float* Xrow = &xs[(m0 + lr) * LDP];
        for (int k = 0; k < Fin; k += 4) {
            int kk = k + 2 * half;  // lanes 16-31 carry K+2,K+3 (ISA A/B layout)
            v2f a, b;
            a.x = Arow[kk];                a.y = Arow[kk + 1];
            b.x = wrel[kk * Fout + n];     b.y = wrel[(kk + 1) * Fout + n];
            c = __builtin_amdgcn_wmma_f32_16x16x4_f32(false, a, false, b,
                                                      (short)0, c, false, false);
            a.x = Xrow[kk];                a.y = Xrow[kk + 1];
            b.x = wroot[kk * Fout + n];    b.y = wroot[(kk + 1) * Fout + n];
            c = __builtin_amdgcn_wmma_f32_16x16x4_f32(false, a, false, b,
                                                      (short)0, c, false, false);
        }
#pragma unroll
        for (int r = 0; r < 8; ++r) {
            int m = m0 + r + 8 * half;  // C layout: VGPR r -> rows r / r+8
            if (m < NPG) {
                float v = c[r] > 0.0f ? c[r] : 0.0f;  // ReLU
                hout[(size_t)(base + m) * Fout + n] = v;
                atomicAdd(&bsum[n], v);
                atomicAdd(&bsq[n], v * v);
            }
        }
    }
    __syncthreads();
    if (tid < Fout) {
        atomicAdd(&stats[tid], bsum[tid]);
        atomicAdd(&stats[64 + tid], bsq[tid]);
    }
}

__global__ void bnfin_kernel(const float* __restrict__ stats, const float* __restrict__ gamma,
                             const float* __restrict__ beta, float* __restrict__ scale,
                             float* __restrict__ shift, int F) {
    int f = threadIdx.x;
    if (f < F) {
        const float invN = 1.0f / (float)N_NODES;
        float m = stats[f] * invN;
        float v = stats[64 + f] * invN - m * m;
        float sc = gamma[f] * rsqrtf(v + 1e-6f);
        scale[f] = sc;
        shift[f] = beta[f] - m * sc;
    }
}

// ------------- fused score GraphConv + tanh + argmax pool + MLP + log_softmax
__global__ __launch_bounds__(64) void score_mlp_kernel(
    const float* __restrict__ h3, const float* __restrict__ sc3,
    const float* __restrict__ sh3, const int* __restrict__ src,
    const int* __restrict__ dst, const int* __restrict__ offs,
    const int* __restrict__ esort, const float* __restrict__ wsrel,
    const float* __restrict__ bsrel, const float* __restrict__ wsroot,
    const float* __restrict__ linw, const float* __restrict__ linb,
    const float* __restrict__ lin1w, const float* __restrict__ lin1b,
    const float* __restrict__ lin2w, const float* __restrict__ lin2b,
    float* __restrict__ out) {
    __shared__ float hs[NPG * 33];
    __shared__ float ags[NPG * 33];
    __shared__ float scv[NPG];
    __shared__ float xp[32], o1[32], o2[16];
    __shared__ int bestIdx;
    const int g = blockIdx.x, base = g * NPG, tid = threadIdx.x;
    const int wave = tid >> 5, lane = tid & 31;

    for (int i = tid; i < NPG * 32; i += 64) {
        int r = i >> 5, c = i & 31;
        hs[r * 33 + c] = h3[(size_t)(base + r) * 32 + c] * sc3[c] + sh3[c];
        ags[r * 33 + c] = 0.0f;
    }
    __syncthreads();
    int e0 = offs[g], e1 = offs[g + 1];
    for (int e = e0 + wave; e < e1; e += 2) {  // unweighted aggregation
        int eid = esort[e];
        int s = src[eid] - base, d = dst[eid] - base;
        atomicAdd(&ags[d * 33 + lane], hs[s * 33 + lane]);
    }
    __syncthreads();
    if (tid < NPG) {
        float acc = bsrel[0];
        for (int f = 0; f < 32; ++f)
            acc += ags[tid * 33 + f] * wsrel[f] + hs[tid * 33 + f] * wsroot[f];
        scv[tid] = tanhf(acc);
    }
    __syncthreads();
    if (tid == 0) {  // first argmax (matches segment_min-of-argmax tie-break)
        float best = scv[0]; int bi = 0;
        for (int i = 1; i < NPG; ++i) if (scv[i] > best) { best = scv[i]; bi = i; }
        bestIdx = bi;
    }
    __syncthreads();
    if (tid < 32) xp[tid] = hs[bestIdx * 33 + tid] * scv[bestIdx];
    __syncthreads();
    if (tid < 32) {  // sort-pool pads with zeros -> rows 32..63 of lin_w unused
        float acc = linb[tid];
        for (int f = 0; f < 32; ++f) acc += xp[f] * linw[f * 32 + tid];
        o1[tid] = acc > 0.0f ? acc : 0.0f;
    }
    __syncthreads();
    if (tid < 16) {
        float acc = lin1b[tid];
        for (int j = 0; j < 32; ++j) acc += o1[j] * lin1w[j * 16 + tid];
        o2[tid] = acc;
    }
    __syncthreads();
    if (tid == 0) {
        float a = lin2b[0], b = lin2b[1];
        for (int j = 0; j < 16; ++j) { a += o2[j] * lin2w[j * 2]; b += o2[j] * lin2w[j * 2 + 1]; }
        float m = a > b ? a : b;
        float lse = m + logf(expf(a - m) + expf(b - m));
        out[g * 2 + 0] = a - lse;
        out[g * 2 + 1] = b - lse;
    }
}

// ---------------------------------------------------------------------- host
extern "C" void kernel_launch(void* const* d_in, const int* in_sizes, int n_in,
                              void* d_out, int out_size, void* d_ws, size_t ws_size,
                              hipStream_t stream) {
    const float* x      = (const float*)d_in[0];
    const int*   ei     = (const int*)d_in[1];
    const float* ew     = (const float*)d_in[2];
    const float* w1rel  = (const float*)d_in[4];
    const float* b1rel  = (const float*)d_in[5];
    const float* w1root = (const float*)d_in[6];
    const float* bn1g   = (const float*)d_in[7];
    const float* bn1b   = (const float*)d_in[8];
    const float* w2rel  = (const float*)d_in[9];
    const float* b2rel  = (const float*)d_in[10];
    const float* w2root = (const float*)d_in[11];
    const float* bn2g   = (const float*)d_in[12];
    const float* bn2b   = (const float*)d_in[13];
    const float* w3rel  = (const float*)d_in[14];
    const float* b3rel  = (const float*)d_in[15];
    const float* w3root = (const float*)d_in[16];
    const float* bn3g   = (const float*)d_in[17];
    const float* bn3b   = (const float*)d_in[18];
    const float* wsrel  = (const float*)d_in[19];
    const float* bsrel  = (const float*)d_in[20];
    const float* wsroot = (const float*)d_in[21];
    const float* linw   = (const float*)d_in[22];
    const float* linb   = (const float*)d_in[23];
    const float* lin1w  = (const float*)d_in[24];
    const float* lin1b  = (const float*)d_in[25];
    const float* lin2w  = (const float*)d_in[26];
    const float* lin2b  = (const float*)d_in[27];

    const int E = in_sizes[1] / 2;
    const int* src = ei;
    const int* dst = ei + E;

    // workspace carve-up (256B aligned)
    char* ws = (char*)d_ws;
    size_t o = 0;
    auto carve = [&](size_t bytes) {
        void* p = ws + o;
        o = (o + bytes + 255) & ~(size_t)255;
        return p;
    };
    int*   esort  = (int*)carve((size_t)E * 4);
    int*   counts = (int*)carve(N_GRAPHS * 4);
    int*   offs   = (int*)carve((N_GRAPHS + 1) * 4);
    int*   cursor = (int*)carve(N_GRAPHS * 4);
    float* stats  = (float*)carve(3 * 128 * 4);   // per layer: sum[64] | sq[64]
    float* scale  = (float*)carve(3 * 64 * 4);
    float* shift  = (float*)carve(3 * 64 * 4);
    float* hA     = (float*)carve((size_t)N_NODES * 64 * 4);  // h1, reused for h3
    float* hB     = (float*)carve((size_t)N_NODES * 64 * 4);  // h2

    const int eblk = (E + 255) / 256;

    zero_kernel<<<(N_GRAPHS + 255) / 256, 256, 0, stream>>>(counts, cursor, stats);
    count_kernel<<<eblk, 256, 0, stream>>>(src, counts, E);
    scan_kernel<<<1, 1024, 0, stream>>>(counts, offs);
    scatter_kernel<<<eblk, 256, 0, stream>>>(src, offs, cursor, esort, E);

    layer_kernel<128, 64, false><<<N_GRAPHS, THREADS, 0, stream>>>(
        x, nullptr, nullptr, src, dst, ew, offs, esort, w1rel, b1rel, w1root, hA, stats);
    bnfin_kernel<<<1, 64, 0, stream>>>(stats, bn1g, bn1b, scale, shift, 64);

    layer_kernel<64, 64, true><<<N_GRAPHS, THREADS, 0, stream>>>(
        hA, scale, shift, src, dst, ew, offs, esort, w2rel, b2rel, w2root, hB, stats + 128);
    bnfin_kernel<<<1, 64, 0, stream>>>(stats + 128, bn2g, bn2b, scale + 64, shift + 64, 64);

    layer_kernel<64, 32, true><<<N_GRAPHS, THREADS, 0, stream>>>(
        hB, scale + 64, shift + 64, src, dst, ew, offs, esort, w3rel, b3rel, w3root, hA,
        stats + 256);
    bnfin_kernel<<<1, 64, 0, stream>>>(stats + 256, bn3g, bn3b, scale + 128, shift + 128, 32);

    score_mlp_kernel<<<N_GRAPHS, 64, 0, stream>>>(
        hA, scale + 128, shift + 128, src, dst, offs, esort, wsrel, bsrel, wsroot,
        linw, linb, lin1w, lin1b, lin2w, lin2b, (float*)d_out);
}